// GNNEncoder_75015898792666
// MI455X (gfx1250) — compile-verified
//
#include <hip/hip_runtime.h>
#include <hip/hip_bf16.h>

// GraphSAGE 2-layer GNN for MI455X (gfx1250, wave32).
//
// Roofline: edge scatter dominates (≈3.3 GB of L2-resident gather+atomic
// traffic; x/h/agg all fit in 192MB L2). GEMMs are only 13 GFLOP total, so
// we keep them in full fp32 precision using V_WMMA_F32_16X16X4_F32.

typedef __attribute__((ext_vector_type(2))) float v2f;
typedef __attribute__((ext_vector_type(8))) float v8f;

#define DFEAT 128

// ---------------------------------------------------------------- zero fill
__global__ void zero_f4(float4* __restrict__ p, long n4) {
    long i = (long)blockIdx.x * blockDim.x + threadIdx.x;
    long stride = (long)gridDim.x * blockDim.x;
    float4 z = make_float4(0.f, 0.f, 0.f, 0.f);
    for (; i < n4; i += stride) p[i] = z;
}

// ---------------------------------------------------------------- degrees
__global__ void deg_kernel(const int* __restrict__ dst, float* __restrict__ deg,
                           int n_edges) {
    int e = blockIdx.x * blockDim.x + threadIdx.x;
    if (e < n_edges) atomicAdd(&deg[dst[e]], 1.0f);
}

// ------------------------------------------------- edge gather + scatter-add
// One wave32 per edge: lane L moves x[src][4L..4L+3] into agg[dst][...]
// via 4 global_atomic_add_f32 (deterministic up to FP add order, same as
// segment_sum on GPU backends).
__global__ __launch_bounds__(256)
void sage_scatter(const float* __restrict__ xin, const int* __restrict__ src,
                  const int* __restrict__ dst, float* __restrict__ agg,
                  int n_edges) {
    int wid  = (blockIdx.x * blockDim.x + threadIdx.x) >> 5;
    int lane = threadIdx.x & 31;
    if (wid >= n_edges) return;
    int s = src[wid];
    int d = dst[wid];
    float4 v = ((const float4*)(xin + (size_t)s * DFEAT))[lane];
    float* ar = agg + (size_t)d * DFEAT + lane * 4;
    atomicAdd(ar + 0, v.x);
    atomicAdd(ar + 1, v.y);
    atomicAdd(ar + 2, v.z);
    atomicAdd(ar + 3, v.w);
}

// ------------------------------------------------------ fused SAGE GEMM
// out[16-node tile] = (agg/deg) @ Wl^T + bl + xin @ Wr^T  (optional ReLU)
// One wave per 16-node tile; 8 v8f accumulators cover all 128 out channels.
// A(16x4 f32) per-lane: lane = m+16h holds A[m][2h], A[m][2h+1].
// B(4x16 f32) per-lane: lane = n+16h holds B[2h][n], B[2h+1][n] = W[o][k].
// C/D(16x16): VGPR r, lane n+16h -> row r+8h, col n.
__global__ __launch_bounds__(64)
void sage_gemm(const float* __restrict__ agg, const float* __restrict__ deg,
               const float* __restrict__ xin,
               const float* __restrict__ Wl, const float* __restrict__ bl,
               const float* __restrict__ Wr, float* __restrict__ out,
               int n_tiles, int do_relu) {
    int tile = ((blockIdx.x * blockDim.x + threadIdx.x) >> 5);
    if (tile >= n_tiles) return;              // never taken with exact launch
    int lane = threadIdx.x & 31;
    int m = lane & 15;                        // row within tile (A) / col (B,D)
    int h = lane >> 4;                        // K/M half selector

    int row = tile * 16 + m;
    float rd = 1.0f / fmaxf(deg[row], 1.0f);

    const float* aggRow = agg + (size_t)row * DFEAT;
    const float* xRow   = xin + (size_t)row * DFEAT;
    const float* wlBase = Wl + (size_t)m * DFEAT;   // W row o = t*16 + m
    const float* wrBase = Wr + (size_t)m * DFEAT;

    v8f acc[8] = {};

    for (int k = 0; k < DFEAT; k += 4) {
        int kk = k + 2 * h;
        v2f a_m = *(const v2f*)(aggRow + kk);
        a_m[0] *= rd;
        a_m[1] *= rd;
        v2f a_x = *(const v2f*)(xRow + kk);
#pragma unroll
        for (int t = 0; t < 8; ++t) {
            v2f b_l = *(const v2f*)(wlBase + (size_t)t * 16 * DFEAT + kk);
            v2f b_r = *(const v2f*)(wrBase + (size_t)t * 16 * DFEAT + kk);
            acc[t] = __builtin_amdgcn_wmma_f32_16x16x4_f32(
                false, a_m, false, b_l, (short)0, acc[t], false, false);
            acc[t] = __builtin_amdgcn_wmma_f32_16x16x4_f32(
                false, a_x, false, b_r, (short)0, acc[t], false, false);
        }
    }

#pragma unroll
    for (int t = 0; t < 8; ++t) {
        float bias = bl[t * 16 + m];
        float* orow = out + (size_t)(tile * 16 + 8 * h) * DFEAT + t * 16 + m;
#pragma unroll
        for (int r = 0; r < 8; ++r) {
            float v = acc[t][r] + bias;
            if (do_relu) v = fmaxf(v, 0.0f);
            orow[(size_t)r * DFEAT] = v;
        }
    }
}

// ---------------------------------------------------------------- launcher
extern "C" void kernel_launch(void* const* d_in, const int* in_sizes, int n_in,
                              void* d_out, int out_size, void* d_ws, size_t ws_size,
                              hipStream_t stream) {
    const float* x   = (const float*)d_in[0];
    const int*   ei  = (const int*)d_in[1];     // edge_index [2][E]
    const float* Wl1 = (const float*)d_in[2];
    const float* bl1 = (const float*)d_in[3];
    const float* Wr1 = (const float*)d_in[4];
    const float* Wl2 = (const float*)d_in[5];
    const float* bl2 = (const float*)d_in[6];
    const float* Wr2 = (const float*)d_in[7];
    float* outp = (float*)d_out;

    int n = in_sizes[0] / DFEAT;     // 100000 nodes
    int E = in_sizes[1] / 2;         // 1600000 edges
    const int* src = ei;
    const int* dst = ei + E;

    // workspace layout (floats): deg[n] | agg[n*128] | h[n*128]
    float* deg = (float*)d_ws;
    float* agg = deg + n;
    float* hbuf = agg + (size_t)n * DFEAT;

    int tiles = (n + 15) / 16;                     // 6250 (exact)
    int gemm_blocks = (tiles + 1) / 2;             // 2 waves/block of 64 thr
    int scat_blocks = (int)(((long)E * 32 + 255) / 256);

    // ---- layer 1 ----
    zero_f4<<<2048, 256, 0, stream>>>((float4*)deg, ((long)n * (DFEAT + 1)) / 4);
    deg_kernel<<<(E + 255) / 256, 256, 0, stream>>>(dst, deg, E);
    sage_scatter<<<scat_blocks, 256, 0, stream>>>(x, src, dst, agg, E);
    sage_gemm<<<gemm_blocks, 64, 0, stream>>>(agg, deg, x, Wl1, bl1, Wr1, hbuf,
                                              tiles, 1);
    // ---- layer 2 ----
    zero_f4<<<2048, 256, 0, stream>>>((float4*)agg, ((long)n * DFEAT) / 4);
    sage_scatter<<<scat_blocks, 256, 0, stream>>>(hbuf, src, dst, agg, E);
    sage_gemm<<<gemm_blocks, 64, 0, stream>>>(agg, deg, hbuf, Wl2, bl2, Wr2,
                                              outp, tiles, 0);
}